// LI_66958540144977
// MI455X (gfx1250) — compile-verified
//
#include <hip/hip_runtime.h>

// Leaky-integrator: out[b,t,o] = scan(alpha[o]*u + (1-alpha[o])*(inputs[b,t,:]·weight[o,:] + bias[o]))
// Phase 1: fp32 WMMA GEMM -> d_out (acts as `current`)
// Phase 2: chunked parallel scan (local -> prefix -> apply), apply is in-place on d_out.

constexpr int B   = 16;
constexpr int T   = 4096;
constexpr int IN  = 512;
constexpr int OUT = 512;
constexpr int M   = B * T;          // 65536 rows

constexpr int CH  = 64;             // scan chunk length
constexpr int NCH = T / CH;         // 64 chunks

typedef float v2f __attribute__((ext_vector_type(2)));
typedef float v4f __attribute__((ext_vector_type(4)));
typedef float v8f __attribute__((ext_vector_type(8)));

// ---------------------------------------------------------------------------
// Phase 1: C[m, o] = A[m, :] . W[o, :] + bias[o]      (fp32 WMMA 16x16x4)
// block tile: 128(M) x 128(N), K staged 32 at a time through LDS.
// 8 waves arranged 4(M) x 2(N); each wave computes a 32x64 slab = 2x4 tiles.
// ---------------------------------------------------------------------------
__global__ __launch_bounds__(256)
void li_gemm_wmma(const float* __restrict__ A,     // [M, IN]
                  const float* __restrict__ W,     // [OUT, IN]
                  const float* __restrict__ bias,  // [OUT]
                  float* __restrict__ C)           // [M, OUT]
{
    constexpr int KSTEP = 32;
    constexpr int MT    = 128;
    constexpr int LDSS  = 36;   // row stride (floats): conflict-free & 16B aligned

    __shared__ float Alds[MT * LDSS];   // [m][k]
    __shared__ float Blds[MT * LDSS];   // [n][k]  (== weight's native layout)

    const int tid    = threadIdx.x;
    const int lane   = tid & 31;
    const int w      = tid >> 5;
    const int wm     = w & 3;      // 0..3  -> M offset wm*32
    const int wn     = w >> 2;     // 0..1  -> N offset wn*64
    const int l16    = lane & 15;
    const int half   = lane >> 4;  // lane half selects K pair {0,1} vs {2,3}
    const int Mblock = blockIdx.x * MT;
    const int Nblock = blockIdx.y * MT;

    v8f acc[2][4];
    const v8f vzero = {0.f,0.f,0.f,0.f,0.f,0.f,0.f,0.f};
#pragma unroll
    for (int tm = 0; tm < 2; ++tm)
#pragma unroll
        for (int tn = 0; tn < 4; ++tn) acc[tm][tn] = vzero;

    for (int kb = 0; kb < IN; kb += KSTEP) {
        __syncthreads();
        // Stage tiles: 128 rows x 32 floats each for A and B.
        // 256 threads * 4 iters * float4 = 4096 floats per tile.
#pragma unroll
        for (int it = 0; it < 4; ++it) {
            const int lin = tid + it * 256;
            const int row = lin >> 3;          // 0..127
            const int col = (lin & 7) << 2;    // 0,4,...,28
            v4f av = *(const v4f*)&A[(size_t)(Mblock + row) * IN + kb + col];
            *(v4f*)&Alds[row * LDSS + col] = av;
            v4f bv = *(const v4f*)&W[(size_t)(Nblock + row) * IN + kb + col];
            *(v4f*)&Blds[row * LDSS + col] = bv;
        }
        __syncthreads();

#pragma unroll
        for (int kk = 0; kk < KSTEP; kk += 4) {
            const int col = kk + half * 2;     // K pair per lane half
            v2f af[2], bf[4];
#pragma unroll
            for (int tm = 0; tm < 2; ++tm)
                af[tm] = *(const v2f*)&Alds[(wm * 32 + tm * 16 + l16) * LDSS + col];
#pragma unroll
            for (int tn = 0; tn < 4; ++tn)
                bf[tn] = *(const v2f*)&Blds[(wn * 64 + tn * 16 + l16) * LDSS + col];
#pragma unroll
            for (int tm = 0; tm < 2; ++tm)
#pragma unroll
                for (int tn = 0; tn < 4; ++tn)
                    acc[tm][tn] = __builtin_amdgcn_wmma_f32_16x16x4_f32(
                        /*neg_a=*/false, af[tm], /*neg_b=*/false, bf[tn],
                        /*c_mod=*/(short)0, acc[tm][tn],
                        /*reuse_a=*/false, /*reuse_b=*/false);
        }
    }

    // Epilogue: C/D layout: VGPR r -> (M = r + 8*half, N = l16)
#pragma unroll
    for (int tn = 0; tn < 4; ++tn) {
        const int colg = Nblock + wn * 64 + tn * 16 + l16;
        const float bv = bias[colg];
#pragma unroll
        for (int tm = 0; tm < 2; ++tm) {
            const int rowbase = Mblock + wm * 32 + tm * 16 + half * 8;
#pragma unroll
            for (int r = 0; r < 8; ++r)
                C[(size_t)(rowbase + r) * OUT + colg] = acc[tm][tn][r] + bv;
        }
    }
}

// ---------------------------------------------------------------------------
// Phase 2a: per-chunk local end-state with zero init:
//   s[b,c,o] = sum_{j<CH} a^(CH-1-j) * (1-a) * cur[b, c*CH+j, o]
// ---------------------------------------------------------------------------
__global__ __launch_bounds__(512)
void li_scan_local(const float* __restrict__ cur,
                   const float* __restrict__ alpha,
                   float* __restrict__ s)
{
    const int o = threadIdx.x;
    const int c = blockIdx.x;
    const int b = blockIdx.y;
    const float a   = alpha[o];
    const float oma = 1.f - a;
    size_t base = ((size_t)b * T + (size_t)c * CH) * OUT + o;
    float u = 0.f;
#pragma unroll 8
    for (int j = 0; j < CH; ++j)
        u = fmaf(a, u, oma * cur[base + (size_t)j * OUT]);
    s[((size_t)b * NCH + c) * OUT + o] = u;
}

// ---------------------------------------------------------------------------
// Phase 2b: sequential prefix over chunks: uinit[c] = incoming state of chunk c
//   uinit[0] = 0 ; uinit[c] = a^CH * uinit[c-1] + s[c-1]
// ---------------------------------------------------------------------------
__global__ __launch_bounds__(512)
void li_scan_prefix(const float* __restrict__ s,
                    const float* __restrict__ alpha,
                    float* __restrict__ uinit)
{
    const int o = threadIdx.x;
    const int b = blockIdx.x;
    float aL = alpha[o];
#pragma unroll
    for (int i = 0; i < 6; ++i) aL *= aL;   // alpha^64 (CH = 2^6)
    float u = 0.f;
    for (int c = 0; c < NCH; ++c) {
        const size_t idx = ((size_t)b * NCH + c) * OUT + o;
        uinit[idx] = u;
        u = fmaf(aL, u, s[idx]);
    }
}

// ---------------------------------------------------------------------------
// Phase 2c: apply scan in-place (each element read then overwritten by the
// same thread, so in-place is safe).
// ---------------------------------------------------------------------------
__global__ __launch_bounds__(512)
void li_scan_apply(float* __restrict__ cur,
                   const float* __restrict__ alpha,
                   const float* __restrict__ uinit)
{
    const int o = threadIdx.x;
    const int c = blockIdx.x;
    const int b = blockIdx.y;
    const float a   = alpha[o];
    const float oma = 1.f - a;
    float u = uinit[((size_t)b * NCH + c) * OUT + o];
    size_t base = ((size_t)b * T + (size_t)c * CH) * OUT + o;
#pragma unroll 8
    for (int j = 0; j < CH; ++j) {
        const size_t idx = base + (size_t)j * OUT;
        u = fmaf(a, u, oma * cur[idx]);
        cur[idx] = u;
    }
}

extern "C" void kernel_launch(void* const* d_in, const int* in_sizes, int n_in,
                              void* d_out, int out_size, void* d_ws, size_t ws_size,
                              hipStream_t stream) {
    const float* inputs = (const float*)d_in[0];   // [B,T,IN]
    const float* weight = (const float*)d_in[1];   // [OUT,IN]
    const float* bias   = (const float*)d_in[2];   // [OUT]
    const float* alpha  = (const float*)d_in[3];   // [OUT]
    float* out = (float*)d_out;                    // [B,T,OUT]

    float* s     = (float*)d_ws;                   // [B,NCH,OUT] = 2 MB
    float* uinit = s + (size_t)B * NCH * OUT;      // [B,NCH,OUT] = 2 MB

    dim3 g1(M / 128, OUT / 128);                   // 512 x 4 blocks
    li_gemm_wmma<<<g1, 256, 0, stream>>>(inputs, weight, bias, out);

    dim3 g2(NCH, B);                               // 64 x 16 blocks
    li_scan_local <<<g2, OUT, 0, stream>>>(out, alpha, s);
    li_scan_prefix<<<B,  OUT, 0, stream>>>(s, alpha, uinit);
    li_scan_apply <<<g2, OUT, 0, stream>>>(out, alpha, uinit);
}